// GraphConv_18537078850015
// MI455X (gfx1250) — compile-verified
//
#include <hip/hip_runtime.h>
#include <hip/hip_bf16.h>

// GraphConv (GCN layer) for MI455X / gfx1250, wave32 + WMMA fp32.
//
//   deg  = segment_sum(1, dst);  norm = clip(deg,1)^-0.5
//   h    = (feat * norm) @ W               [N,256]x[256,128] -> fp32 WMMA
//   rst  = segment_sum(h[src], dst)        L2-resident gather + f32 atomics
//   out  = rst * norm + bias

typedef float v2f __attribute__((ext_vector_type(2)));
typedef float v8f __attribute__((ext_vector_type(8)));

#define IN_F  256
#define OUT_F 128

// ---------------------------------------------------------------- zero fill
__global__ void zero_f32_kernel(float* __restrict__ p, long long n) {
  long long i = (long long)blockIdx.x * blockDim.x + threadIdx.x;
  long long stride = (long long)gridDim.x * blockDim.x;
  for (; i < n; i += stride) p[i] = 0.0f;
}

__global__ void zero_u32_kernel(unsigned* __restrict__ p, int n) {
  int i = blockIdx.x * blockDim.x + threadIdx.x;
  if (i < n) p[i] = 0u;
}

// ------------------------------------------------------------- in-degree
__global__ void degree_kernel(const int* __restrict__ dst,
                              unsigned* __restrict__ deg, int n_edges) {
  int i = blockIdx.x * blockDim.x + threadIdx.x;
  if (i < n_edges) atomicAdd(&deg[dst[i]], 1u);
}

// -------------------------------------------------- norm = clip(deg,1)^-0.5
__global__ void norm_kernel(const unsigned* __restrict__ deg,
                            float* __restrict__ norm, int n_nodes) {
  int i = blockIdx.x * blockDim.x + threadIdx.x;
  if (i < n_nodes) {
    float d = (float)deg[i];
    if (d < 1.0f) d = 1.0f;
    norm[i] = 1.0f / sqrtf(d);
  }
}

// ------------------------------------------- Wt[n][k] = W[k][n]  (fp32)
__global__ void transpose_w_kernel(const float* __restrict__ w,
                                   float* __restrict__ wt) {
  int i = blockIdx.x * blockDim.x + threadIdx.x;  // 256*128 threads
  if (i < IN_F * OUT_F) {
    int k = i / OUT_F;
    int n = i % OUT_F;
    wt[n * IN_F + k] = w[i];
  }
}

// --------------------------------------------------------------- WMMA GEMM
// block = 256 threads = 8 waves. Block owns 16 rows of feat; wave w owns
// output columns [16w, 16w+16). K=256 -> 64 x V_WMMA_F32_16X16X4_F32.
//
// A (16x4 f32) VGPR layout (ISA 7.12.2): lane L, M = L%16,
//   a[0] -> K = (L>=16 ? 2 : 0),  a[1] -> K = (L>=16 ? 3 : 1)
// B (4x16 f32): lane L, N = L%16, b[j] -> K = j + (L>=16 ? 2 : 0)
// C/D (16x16 f32): acc[r]: lanes 0-15 -> M=r, lanes 16-31 -> M=r+8, N=L%16
__global__ void __launch_bounds__(256)
gemm_wmma_kernel(const float* __restrict__ feat,
                 const float* __restrict__ wt,     // [128][256] transposed
                 const float* __restrict__ norm,
                 float* __restrict__ h, int n_nodes) {
  // +4 pad floats per row: bank = (m*260 + k) % 64 -> 4*m spread, no conflicts
  __shared__ float ldsA[16][IN_F + 4];

  const int tid  = threadIdx.x;
  const int tile = blockIdx.x;

  // ---- stage 16 rows of feat, pre-scaled by norm, into LDS ----
  {
    int r   = tid >> 4;          // 0..15 row in tile
    int seg = tid & 15;          // 16-float segment of the row
    int grow = tile * 16 + r;
    float scale = 0.0f;
    float4 v0 = {0, 0, 0, 0}, v1 = v0, v2 = v0, v3 = v0;
    if (grow < n_nodes) {
      scale = norm[grow];
      const float4* f4 =
          (const float4*)(feat + (long long)grow * IN_F + seg * 16);
      v0 = f4[0]; v1 = f4[1]; v2 = f4[2]; v3 = f4[3];
    }
    float* la = &ldsA[r][seg * 16];
    la[0]  = v0.x * scale; la[1]  = v0.y * scale;
    la[2]  = v0.z * scale; la[3]  = v0.w * scale;
    la[4]  = v1.x * scale; la[5]  = v1.y * scale;
    la[6]  = v1.z * scale; la[7]  = v1.w * scale;
    la[8]  = v2.x * scale; la[9]  = v2.y * scale;
    la[10] = v2.z * scale; la[11] = v2.w * scale;
    la[12] = v3.x * scale; la[13] = v3.y * scale;
    la[14] = v3.z * scale; la[15] = v3.w * scale;
  }
  __syncthreads();

  // ---- WMMA main loop ----
  const int lane = tid & 31;
  const int w    = tid >> 5;           // wave id 0..7
  const int m    = lane & 15;          // M (for A) / N (for B,C)
  const int klo  = (lane >> 4) << 1;   // K sub-offset: 0 (lanes 0-15) or 2

  const float* arow = &ldsA[m][0];
  const float* brow = wt + (w * 16 + m) * IN_F;  // Wt row = output column

  v8f acc = {};
#pragma unroll
  for (int kk = 0; kk < IN_F / 4; ++kk) {
    int kb = kk * 4 + klo;
    v2f a = *(const v2f*)(arow + kb);   // ds_load_b64
    v2f b = *(const v2f*)(brow + kb);   // global_load_b64 (L2-resident)
    acc = __builtin_amdgcn_wmma_f32_16x16x4_f32(
        /*neg_a=*/false, a, /*neg_b=*/false, b,
        /*c_mod=*/(short)0, acc, /*reuse_a=*/false, /*reuse_b=*/false);
  }

  // ---- write D tile to h ----
  const int colg = w * 16 + m;
  const int mhi  = (lane >> 4) << 3;   // +8 rows for lanes 16-31
#pragma unroll
  for (int rr = 0; rr < 8; ++rr) {
    int rowg = tile * 16 + rr + mhi;
    if (rowg < n_nodes) h[(long long)rowg * OUT_F + colg] = acc[rr];
  }
}

// ----------------------------------------------------------- edge scatter
// One wave per edge: lane moves float4 -> 512B coalesced gather from h[src]
// (L2-resident, 51 MB < 192 MB L2) + 4 fp32 atomic adds into out[dst].
__global__ void __launch_bounds__(256)
scatter_kernel(const float* __restrict__ h, const int* __restrict__ src,
               const int* __restrict__ dst, float* __restrict__ out,
               int n_edges) {
  int edge = blockIdx.x * 8 + (threadIdx.x >> 5);
  int lane = threadIdx.x & 31;
  if (edge >= n_edges) return;
  int s = src[edge];
  int d = dst[edge];
  float4 v = ((const float4*)(h + (long long)s * OUT_F))[lane];
  float* od = out + (long long)d * OUT_F + lane * 4;
  atomicAdd(od + 0, v.x);
  atomicAdd(od + 1, v.y);
  atomicAdd(od + 2, v.z);
  atomicAdd(od + 3, v.w);
}

// -------------------------------------------------- out = out*norm + bias
__global__ void epilogue_kernel(float* __restrict__ out,
                                const float* __restrict__ norm,
                                const float* __restrict__ bias,
                                long long total) {
  long long i = (long long)blockIdx.x * blockDim.x + threadIdx.x;
  if (i < total) {
    int node = (int)(i >> 7);     // /128
    int c    = (int)(i & 127);    // %128
    out[i] = out[i] * norm[node] + bias[c];
  }
}

extern "C" void kernel_launch(void* const* d_in, const int* in_sizes, int n_in,
                              void* d_out, int out_size, void* d_ws,
                              size_t ws_size, hipStream_t stream) {
  const float* feat   = (const float*)d_in[0];
  const float* weight = (const float*)d_in[1];
  const float* bias   = (const float*)d_in[2];
  const int*   src    = (const int*)d_in[3];
  const int*   dst    = (const int*)d_in[4];
  float* out = (float*)d_out;

  const int n_nodes = in_sizes[0] / IN_F;
  const int n_edges = in_sizes[3];

  // workspace layout (fp32 words): h[N*128] | Wt[128*256] | norm[N] | deg[N]
  float*    h    = (float*)d_ws;
  float*    wt   = h + (size_t)n_nodes * OUT_F;
  float*    norm = wt + (size_t)IN_F * OUT_F;
  unsigned* deg  = (unsigned*)(norm + n_nodes);

  const long long total_out = (long long)n_nodes * OUT_F;

  // 1. zero accumulators (fresh every call: graph-replay safe)
  zero_f32_kernel<<<4096, 256, 0, stream>>>(out, total_out);
  zero_u32_kernel<<<(n_nodes + 255) / 256, 256, 0, stream>>>(deg, n_nodes);

  // 2. in-degree via exact integer atomics
  degree_kernel<<<(n_edges + 255) / 256, 256, 0, stream>>>(dst, deg, n_edges);

  // 3. norm = clip(deg,1)^-0.5
  norm_kernel<<<(n_nodes + 255) / 256, 256, 0, stream>>>(deg, norm, n_nodes);

  // 4. transpose weight -> Wt[128][256] for contiguous per-lane B loads
  transpose_w_kernel<<<(IN_F * OUT_F + 255) / 256, 256, 0, stream>>>(weight,
                                                                     wt);

  // 5. h = (feat * norm) @ W  via V_WMMA_F32_16X16X4_F32
  gemm_wmma_kernel<<<(n_nodes + 15) / 16, 256, 0, stream>>>(feat, wt, norm, h,
                                                            n_nodes);

  // 6. rst = segment_sum(h[src], dst)  (one wave per edge)
  scatter_kernel<<<(n_edges + 7) / 8, 256, 0, stream>>>(h, src, dst, out,
                                                        n_edges);

  // 7. out = out * norm + bias
  epilogue_kernel<<<(int)((total_out + 255) / 256), 256, 0, stream>>>(
      out, norm, bias, total_out);
}